// VoxelResBackBone8xEDFDNet_44092134261309
// MI455X (gfx1250) — compile-verified
//
#include <hip/hip_runtime.h>

#define CIN   5
#define COUT  16
#define KOFF  27
#define KTOT  (KOFF * CIN)          // 135
#define KCHUNKS ((KTOT + 3) / 4)    // 34 WMMA k-steps of 4
#define BN_EPS 1e-3f
#define WPB   8                     // waves (16-voxel tiles) per 256-thread block

typedef float v2f __attribute__((ext_vector_type(2)));
typedef float v8f __attribute__((ext_vector_type(8)));

// ---------------- kernel 0: zero output + stat accumulators ----------------
__global__ void zero_kernel(float* __restrict__ out, int n_out,
                            float* __restrict__ stats /* 32 floats */) {
    int i = blockIdx.x * blockDim.x + threadIdx.x;
    if (i < n_out) out[i] = 0.f;
    if (i < 32)    stats[i] = 0.f;
}

// ---------------- kernel 1: rulebook gather -> f32 WMMA GEMM + BN stats ----
__global__ __launch_bounds__(256) void subm_conv_wmma_kernel(
    const float* __restrict__ feat,    // (N, 5)
    const float* __restrict__ conv_w,  // (135, 16) row-major == (27,5,16)
    const int*   __restrict__ in_idx,  // (N, 27)
    float*       __restrict__ out_pre, // (N, 16)
    float*       __restrict__ g_sum,   // (16)
    float*       __restrict__ g_sumsq, // (16)
    int N)
{
    __shared__ float s_w[KTOT * COUT];        // 8.6 KB: whole weight matrix
    __shared__ float s_sum[COUT];
    __shared__ float s_sq[COUT];

    const int tid  = threadIdx.x;
    const int w    = tid >> 5;
    const int lane = tid & 31;

    // stage weights (block-cooperative) and zero block stat accumulators
    for (int t = tid; t < KTOT * COUT; t += 256) s_w[t] = conv_w[t];
    if (tid < COUT) { s_sum[tid] = 0.f; s_sq[tid] = 0.f; }
    __syncthreads();

    const int  half = lane >> 4;   // lanes 16-31 handle k+2 / rows m+8
    const int  m    = lane & 15;   // A row index == B column index
    const bool hlf  = (half != 0);

    // ---- preload this lane's rulebook row into registers (27 VGPRs) ----
    const int tile = blockIdx.x * WPB + w;
    const int base = tile * 16;
    const int vm   = base + m;
    const int vrow = (vm < N) ? vm : (N - 1);    // clamp; out-of-range waves discard results
    const int* __restrict__ rowp = in_idx + (long)vrow * KOFF;
    int idxreg[KOFF];
    #pragma unroll
    for (int ko = 0; ko < KOFF; ++ko) idxreg[ko] = rowp[ko];

    // weight pointer: per-lane base covers column m and the half k-shift (+2 rows)
    const float* __restrict__ swp = s_w + m + half * (2 * COUT);

    v8f acc = {};
    #pragma unroll
    for (int ch = 0; ch < KCHUNKS; ++ch) {
        v2f a, b;
        #pragma unroll
        for (int v = 0; v < 2; ++v) {
            // compile-time per-iteration constants (full unroll)
            const int  kk0  = ch * 4 + v;               // k for half==0 lanes
            const int  kk1  = kk0 + 2;                  // k for half==1 lanes
            const bool tail = (kk1 >= KTOT);            // only ch==33, v==1
            const int  kk1c = tail ? (KTOT - 1) : kk1;
            const int  ko0  = kk0 / CIN, c0 = kk0 % CIN;
            const int  ko1  = kk1c / CIN, c1 = kk1c % CIN;

            int idx = hlf ? idxreg[ko1] : idxreg[ko0];  // v_cndmask between regs
            if (tail && hlf) idx = -1;                  // pad lane -> no gather
            const int c = hlf ? c1 : c0;                // cndmask of constants

            float av = 0.f;
            if (idx >= 0) av = feat[idx * CIN + c];     // exec-masked: ~96% lanes skip

            float bv = swp[kk0 * COUT];                 // ds_load, constant offset
            if (tail && hlf) bv = 0.f;

            a[v] = av;
            b[v] = bv;
        }
        // D = A(16x4) x B(4x16) + C, exact f32 math on the matrix pipes
        acc = __builtin_amdgcn_wmma_f32_16x16x4_f32(
            /*neg_a=*/false, a, /*neg_b=*/false, b,
            /*c_mod=*/(short)0, acc, /*reuse_a=*/false, /*reuse_b=*/false);
    }

    // C/D layout: VGPR r, lane -> row (r + 8*half), col m
    float s = 0.f, s2 = 0.f;
    #pragma unroll
    for (int r = 0; r < 8; ++r) {
        int vo = base + r + 8 * half;
        if (vo < N) {
            float val = acc[r];
            out_pre[vo * COUT + m] = val;
            s  += val;
            s2 += val * val;
        }
    }
    // per-channel partial sums: LDS reduce, then 16 global atomics per block
    atomicAdd(&s_sum[m], s);
    atomicAdd(&s_sq[m],  s2);
    __syncthreads();
    if (tid < COUT) {
        atomicAdd(&g_sum[tid],   s_sum[tid]);
        atomicAdd(&g_sumsq[tid], s_sq[tid]);
    }
}

// ---------------- kernel 2: BN affine parameters from batch stats ----------
__global__ void bn_params_kernel(const float* __restrict__ g_sum,
                                 const float* __restrict__ g_sumsq,
                                 const float* __restrict__ gamma,
                                 const float* __restrict__ beta,
                                 float* __restrict__ scale,
                                 float* __restrict__ shift, int N) {
    int d = threadIdx.x;
    if (d < COUT) {
        float inv_n = 1.0f / (float)N;
        float mu  = g_sum[d] * inv_n;
        float var = g_sumsq[d] * inv_n - mu * mu;   // biased var, matches jnp.var
        float sc  = gamma[d] * rsqrtf(var + BN_EPS);
        scale[d] = sc;
        shift[d] = beta[d] - mu * sc;
    }
}

// ---------------- kernel 3: BN + ReLU + BEV segment-sum scatter ------------
// one thread = 4 consecutive channels of one voxel (float4 load of out_pre)
__global__ void bn_relu_scatter_kernel(const float4* __restrict__ out_pre4,
                                       const int*   __restrict__ inv_idx,
                                       const float* __restrict__ scale,
                                       const float* __restrict__ shift,
                                       float* __restrict__ bev, int N) {
    int i = blockIdx.x * blockDim.x + threadIdx.x;     // i over N*4 quads
    if (i < N * (COUT / 4)) {
        int n  = i >> 2;                 // voxel
        int dq = (i & 3) * 4;            // channel quad base
        float4 p = out_pre4[i];
        int seg = inv_idx[n] * COUT + dq;
        float v0 = fmaf(p.x, scale[dq + 0], shift[dq + 0]);
        float v1 = fmaf(p.y, scale[dq + 1], shift[dq + 1]);
        float v2 = fmaf(p.z, scale[dq + 2], shift[dq + 2]);
        float v3 = fmaf(p.w, scale[dq + 3], shift[dq + 3]);
        atomicAdd(&bev[seg + 0], v0 > 0.f ? v0 : 0.f);
        atomicAdd(&bev[seg + 1], v1 > 0.f ? v1 : 0.f);
        atomicAdd(&bev[seg + 2], v2 > 0.f ? v2 : 0.f);
        atomicAdd(&bev[seg + 3], v3 > 0.f ? v3 : 0.f);
    }
}

extern "C" void kernel_launch(void* const* d_in, const int* in_sizes, int n_in,
                              void* d_out, int out_size, void* d_ws, size_t ws_size,
                              hipStream_t stream) {
    const float* feat    = (const float*)d_in[0];  // (N,5)
    const float* conv_w  = (const float*)d_in[1];  // (27,5,16) == (135,16)
    const float* gamma   = (const float*)d_in[2];
    const float* beta    = (const float*)d_in[3];
    const int*   in_idx  = (const int*)  d_in[4];  // (N,27)
    const int*   inv_idx = (const int*)  d_in[5];  // (N,)
    const int N = in_sizes[0] / CIN;

    // workspace layout
    float* ws_out  = (float*)d_ws;                  // N*16 pre-BN activations
    float* g_sum   = ws_out + (size_t)N * COUT;     // 16
    float* g_sumsq = g_sum + COUT;                  // 16
    float* scale   = g_sumsq + COUT;                // 16
    float* shift   = scale + COUT;                  // 16
    float* bev     = (float*)d_out;                 // (n_unique, 16)

    int zn = out_size > 32 ? out_size : 32;
    zero_kernel<<<(zn + 255) / 256, 256, 0, stream>>>(bev, out_size, g_sum);

    int tiles  = (N + 15) / 16;
    int blocks = (tiles + WPB - 1) / WPB;
    subm_conv_wmma_kernel<<<blocks, 256, 0, stream>>>(
        feat, conv_w, in_idx, ws_out, g_sum, g_sumsq, N);

    bn_params_kernel<<<1, 32, 0, stream>>>(g_sum, g_sumsq, gamma, beta, scale, shift, N);

    int quads = N * (COUT / 4);
    bn_relu_scatter_kernel<<<(quads + 255) / 256, 256, 0, stream>>>(
        (const float4*)ws_out, inv_idx, scale, shift, bev, N);
}